// SimpleConvRNN_28604482191895
// MI455X (gfx1250) — compile-verified
//
#include <hip/hip_runtime.h>

typedef __attribute__((ext_vector_type(2))) float v2f;
typedef __attribute__((ext_vector_type(8))) float v8f;
typedef unsigned int v4u __attribute__((ext_vector_type(4)));
typedef int v8i __attribute__((ext_vector_type(8)));
typedef int v4i __attribute__((ext_vector_type(4)));

#define HW        6144          // 64*96 pixels per batch image
#define TILES_PB  384           // HW / 16
#define WST       132           // padded LDS row stride (floats) -> bank-conflict-free operand fetch
#define SW_FLOATS   (128*WST)   // one 128x128 weight matrix, padded
#define SW2_FLOATS  (64*WST)    // last layer: only rows 64..127 needed
#define SX_FLOATS   (8*16*WST)  // 8 waves * 16-pixel tile * 128ch (padded)
#define SMEM_FLOATS (2*SW_FLOATS + SW2_FLOATS + 320 + SX_FLOATS)

// ---- Tensor Data Mover availability (ROCm7.2 clang-22: 5 args; clang>=23: 6 args)
#if defined(__has_builtin)
#if __has_builtin(__builtin_amdgcn_tensor_load_to_lds) && \
    __has_builtin(__builtin_amdgcn_s_wait_tensorcnt)
#define USE_TDM 1
#endif
#endif
#ifndef USE_TDM
#define USE_TDM 0
#endif

// ---------------------------------------------------------------- 4x4 inverse
__device__ __forceinline__ void inv4(const float m[16], float o[16]) {
    float inv[16];
    inv[0]  =  m[5]*m[10]*m[15] - m[5]*m[11]*m[14] - m[9]*m[6]*m[15] + m[9]*m[7]*m[14] + m[13]*m[6]*m[11] - m[13]*m[7]*m[10];
    inv[4]  = -m[4]*m[10]*m[15] + m[4]*m[11]*m[14] + m[8]*m[6]*m[15] - m[8]*m[7]*m[14] - m[12]*m[6]*m[11] + m[12]*m[7]*m[10];
    inv[8]  =  m[4]*m[9]*m[15]  - m[4]*m[11]*m[13] - m[8]*m[5]*m[15] + m[8]*m[7]*m[13] + m[12]*m[5]*m[11] - m[12]*m[7]*m[9];
    inv[12] = -m[4]*m[9]*m[14]  + m[4]*m[10]*m[13] + m[8]*m[5]*m[14] - m[8]*m[6]*m[13] - m[12]*m[5]*m[10] + m[12]*m[6]*m[9];
    inv[1]  = -m[1]*m[10]*m[15] + m[1]*m[11]*m[14] + m[9]*m[2]*m[15] - m[9]*m[3]*m[14] - m[13]*m[2]*m[11] + m[13]*m[3]*m[10];
    inv[5]  =  m[0]*m[10]*m[15] - m[0]*m[11]*m[14] - m[8]*m[2]*m[15] + m[8]*m[3]*m[14] + m[12]*m[2]*m[11] - m[12]*m[3]*m[10];
    inv[9]  = -m[0]*m[9]*m[15]  + m[0]*m[11]*m[13] + m[8]*m[1]*m[15] - m[8]*m[3]*m[13] - m[12]*m[1]*m[11] + m[12]*m[3]*m[9];
    inv[13] =  m[0]*m[9]*m[14]  - m[0]*m[10]*m[13] - m[8]*m[1]*m[14] + m[8]*m[2]*m[13] + m[12]*m[1]*m[10] - m[12]*m[2]*m[9];
    inv[2]  =  m[1]*m[6]*m[15]  - m[1]*m[7]*m[14]  - m[5]*m[2]*m[15] + m[5]*m[3]*m[14] + m[13]*m[2]*m[7]  - m[13]*m[3]*m[6];
    inv[6]  = -m[0]*m[6]*m[15]  + m[0]*m[7]*m[14]  + m[4]*m[2]*m[15] - m[4]*m[3]*m[14] - m[12]*m[2]*m[7]  + m[12]*m[3]*m[6];
    inv[10] =  m[0]*m[5]*m[15]  - m[0]*m[7]*m[13]  - m[4]*m[1]*m[15] + m[4]*m[3]*m[13] + m[12]*m[1]*m[7]  - m[12]*m[3]*m[5];
    inv[14] = -m[0]*m[5]*m[14]  + m[0]*m[6]*m[13]  + m[4]*m[1]*m[14] - m[4]*m[2]*m[13] - m[12]*m[1]*m[6]  + m[12]*m[2]*m[5];
    inv[3]  = -m[1]*m[6]*m[11]  + m[1]*m[7]*m[10]  + m[5]*m[2]*m[11] - m[5]*m[3]*m[10] - m[9]*m[2]*m[7]   + m[9]*m[3]*m[6];
    inv[7]  =  m[0]*m[6]*m[11]  - m[0]*m[7]*m[10]  - m[4]*m[2]*m[11] + m[4]*m[3]*m[10] + m[8]*m[2]*m[7]   - m[8]*m[3]*m[6];
    inv[11] = -m[0]*m[5]*m[11]  + m[0]*m[7]*m[9]   + m[4]*m[1]*m[11] - m[4]*m[3]*m[9]  - m[8]*m[1]*m[7]   + m[8]*m[3]*m[5];
    inv[15] =  m[0]*m[5]*m[10]  - m[0]*m[6]*m[9]   - m[4]*m[1]*m[10] + m[4]*m[2]*m[9]  + m[8]*m[1]*m[6]   - m[8]*m[2]*m[5];
    float det = m[0]*inv[0] + m[1]*inv[4] + m[2]*inv[8] + m[3]*inv[12];
    det = 1.0f / det;
    #pragma unroll
    for (int i = 0; i < 16; ++i) o[i] = inv[i] * det;
}

// ------------------------------------------------- per-batch xf = cur @ inv(prev)
__global__ void scrnn_xf_kernel(const float* __restrict__ prev_ext,
                                const float* __restrict__ cur_ext,
                                const int*   __restrict__ midx,
                                const unsigned char* __restrict__ umem,
                                float* __restrict__ xf, int B)
{
    int b = blockIdx.x * blockDim.x + threadIdx.x;
    if (b >= B) return;
    int mi = midx[b];
    float P[16];
    if (umem[b]) {
        #pragma unroll
        for (int i = 0; i < 16; ++i) P[i] = prev_ext[mi * 16 + i];
    } else {
        #pragma unroll
        for (int i = 0; i < 16; ++i) P[i] = (i % 5 == 0) ? 1.0f : 0.0f;
    }
    float Pi[16];
    inv4(P, Pi);
    const float* C = cur_ext + b * 16;
    #pragma unroll
    for (int i = 0; i < 4; ++i)
        #pragma unroll
        for (int j = 0; j < 4; ++j) {
            float s = 0.0f;
            #pragma unroll
            for (int k = 0; k < 4; ++k) s += C[i * 4 + k] * Pi[k * 4 + j];
            xf[b * 16 + i * 4 + j] = s;
        }
}

#if USE_TDM
// ---------------------------------------------------------------- TDM weight DMA
// 1-D tile of `nd` DWORDs, pad 4 DWORDs after every 128 DWORDs stored
// -> produces the WST=132 padded row layout in LDS directly.
__device__ __forceinline__ void tdm_load_padded(unsigned int lds_byte_addr,
                                                const void* gptr,
                                                unsigned int nd)
{
    unsigned long long ga = (unsigned long long)(__SIZE_TYPE__)gptr;
    v4u g0;
    g0.x = 1u;                                                  // count=1, user mode
    g0.y = lds_byte_addr;                                       // lds_addr
    g0.z = (unsigned int)ga;                                    // global_addr[31:0]
    g0.w = (unsigned int)((ga >> 32) & 0x01FFFFFFu) | (2u << 30); // addr[56:32], type=2
    v8i g1;
    g1[0] = (int)((2u << 16)        // data_size = 4B
                | (1u << 20)        // pad_enable
                | (6u << 22)        // pad_interval: every 128 DWORDs
                | (3u << 25));      // pad_amount: 4 DWORDs
    g1[1] = (int)((nd & 0xFFFFu) << 16);                 // tensor_dim0[15:0]
    g1[2] = (int)(((nd >> 16) & 0xFFFFu) | (1u << 16));  // tensor_dim0[31:16], tensor_dim1=1
    g1[3] = (int)((nd & 0xFFFFu) << 16);                 // tile_dim0 = nd
    g1[4] = 0;                                           // tile_dim1=0, tile_dim2=0
    g1[5] = (int)nd;                                     // tensor_dim0_stride[31:0]
    g1[6] = 0;
    g1[7] = 0;
    v4i zz = {0, 0, 0, 0};
#if __clang_major__ >= 23
    v8i zz8 = {0, 0, 0, 0, 0, 0, 0, 0};
    __builtin_amdgcn_tensor_load_to_lds(g0, g1, zz, zz, zz8, 0);
#else
    __builtin_amdgcn_tensor_load_to_lds(g0, g1, zz, zz, 0);
#endif
}
#endif

// ------------------------------------- one fused 128->128 layer (+bias, +relu)
// A operand: lane L holds W[m0 + (L&15)][k0 + 2*(L>>4) .. +1]      (16x4 f32 layout)
// B operand: lane L holds X[pix = L&15][k0 + 2*(L>>4) .. +1]       (4x16 f32 layout)
// D operand: VGPR r = out-ch (m0 + r + 8*(L>>4)), pixel = L&15
__device__ __forceinline__ void fc_relu_128(const float* __restrict__ sW,
                                            const float* __restrict__ sB,
                                            float* __restrict__ myX,
                                            int pr, int hi)
{
    v8f acc[8];
    v8f zero = {0.f, 0.f, 0.f, 0.f, 0.f, 0.f, 0.f, 0.f};
    #pragma unroll
    for (int m = 0; m < 8; ++m) acc[m] = zero;

    #pragma unroll 4
    for (int s = 0; s < 32; ++s) {
        const int col = 4 * s + 2 * hi;
        v2f bv = *(const v2f*)(myX + pr * WST + col);
        #pragma unroll
        for (int m = 0; m < 8; ++m) {
            v2f av = *(const v2f*)(sW + (m * 16 + pr) * WST + col);
            acc[m] = __builtin_amdgcn_wmma_f32_16x16x4_f32(
                false, av, false, bv, (short)0, acc[m], false, false);
        }
    }

    #pragma unroll
    for (int m = 0; m < 8; ++m) {
        #pragma unroll
        for (int r = 0; r < 8; ++r) {
            int ch = m * 16 + r + 8 * hi;
            float v = acc[m][r] + sB[ch];
            myX[pr * WST + ch] = v > 0.0f ? v : 0.0f;
        }
    }
}

// ------------------------- build one 16-pixel x-tile: FTL(mem) | mem | img
__device__ __forceinline__ void build_tile(float* __restrict__ myX, int gt,
                                           const float* __restrict__ img,
                                           const float* __restrict__ mem,
                                           const int*   __restrict__ midx,
                                           const unsigned char* __restrict__ umem,
                                           const float* __restrict__ xf,
                                           int lane)
{
    const int b  = gt / TILES_PB;
    const int pb = (gt % TILES_PB) * 16;
    const int   mi = midx[b];
    const float um = umem[b] ? 1.0f : 0.0f;
    const float* memb = mem + ((size_t)mi * 64) * HW + pb;
    const float* imgb = img + ((size_t)b  * 64) * HW + pb;

    float XF[16];
    #pragma unroll
    for (int i = 0; i < 16; ++i) XF[i] = xf[b * 16 + i];

    for (int idx = lane; idx < 2048; idx += 32) {
        const int ch = idx >> 4;     // lanes 0-15: even ch, 16-31: odd ch
        const int p  = idx & 15;
        float v;
        if (ch < 32) {
            const int g = ch >> 2, i = ch & 3;
            const float* mg = memb + (g * 4) * HW + p;
            v = XF[i*4+0]*mg[0] + XF[i*4+1]*mg[HW] + XF[i*4+2]*mg[2*HW] + XF[i*4+3]*mg[3*HW];
            v *= um;
        } else if (ch < 64) {
            v = memb[ch * HW + p] * um;
        } else {
            v = imgb[(ch - 64) * HW + p];
        }
        myX[p * WST + ch] = v;
    }
}

// ---------------------------------------------------------------- fused main
__global__ __launch_bounds__(256)
void scrnn_fused_kernel(const float* __restrict__ img,
                        const float* __restrict__ mem,
                        const int*   __restrict__ midx,
                        const unsigned char* __restrict__ umem,
                        const float* __restrict__ xf,
                        const float* __restrict__ Ws,
                        const float* __restrict__ bs,
                        float* __restrict__ out)
{
    extern __shared__ float smem[];
    float* sW0 = smem;                    // 128 x 128, row stride WST
    float* sW1 = sW0 + SW_FLOATS;
    float* sW2 = sW1 + SW_FLOATS;         // rows 64..127 of Ws[2]
    float* sB0 = sW2 + SW2_FLOATS;
    float* sB1 = sB0 + 128;
    float* sB2 = sB1 + 128;
    float* sX  = sB2 + 64;

    const int tid  = threadIdx.x;
    const int wv   = tid >> 5;
    const int lane = tid & 31;
    const int pr   = lane & 15;
    const int hi   = lane >> 4;
    float* myX = sX + wv * (16 * WST);    // wave-private 16px x 128ch tile

#if USE_TDM
    // Async weight DMA (global -> padded LDS layout), overlapped with x-tile build.
    if (wv == 0) {
        unsigned int base = (unsigned int)__builtin_amdgcn_groupstaticsize();
        tdm_load_padded(base,                            Ws,         16384u);
        tdm_load_padded(base + SW_FLOATS * 4u,           Ws + 16384, 16384u);
        tdm_load_padded(base + 2u * SW_FLOATS * 4u,      Ws + 40960, 8192u); // Ws[2] rows 64..127
    }
#else
    for (int i = tid * 4; i < 16384; i += 1024) {
        float4 v = *(const float4*)&Ws[i];
        *(float4*)&sW0[(i >> 7) * WST + (i & 127)] = v;
    }
    for (int i = tid * 4; i < 16384; i += 1024) {
        float4 v = *(const float4*)&Ws[16384 + i];
        *(float4*)&sW1[(i >> 7) * WST + (i & 127)] = v;
    }
    for (int i = tid * 4; i < 8192; i += 1024) {
        float4 v = *(const float4*)&Ws[40960 + i];
        *(float4*)&sW2[(i >> 7) * WST + (i & 127)] = v;
    }
#endif
    if (tid < 128) { sB0[tid] = bs[tid]; sB1[tid] = bs[128 + tid]; }
    if (tid < 64)  { sB2[tid] = bs[320 + tid]; }     // bs[2][64:]

    const int gt0 = (blockIdx.x * 8 + wv) * 4;       // this wave's first tile
    build_tile(myX, gt0, img, mem, midx, umem, xf, lane);

#if USE_TDM
    if (wv == 0) __builtin_amdgcn_s_wait_tensorcnt(0);
#endif
    __syncthreads();

    #pragma unroll 1
    for (int tt = 0; tt < 4; ++tt) {
        const int gt = gt0 + tt;
        const int b  = gt / TILES_PB;
        const int pb = (gt % TILES_PB) * 16;

        fc_relu_128(sW0, sB0, myX, pr, hi);   // layer 0
        fc_relu_128(sW1, sB1, myX, pr, hi);   // layer 1

        // Layer 2: only final channels 64..127 are needed -> 4 M-tiles, no relu,
        // results go straight to global memory.
        v8f acc[4];
        v8f zero = {0.f, 0.f, 0.f, 0.f, 0.f, 0.f, 0.f, 0.f};
        #pragma unroll
        for (int m = 0; m < 4; ++m) acc[m] = zero;
        #pragma unroll 4
        for (int s = 0; s < 32; ++s) {
            const int col = 4 * s + 2 * hi;
            v2f bv = *(const v2f*)(myX + pr * WST + col);
            #pragma unroll
            for (int m = 0; m < 4; ++m) {
                v2f av = *(const v2f*)(sW2 + (m * 16 + pr) * WST + col);
                acc[m] = __builtin_amdgcn_wmma_f32_16x16x4_f32(
                    false, av, false, bv, (short)0, acc[m], false, false);
            }
        }
        float* outb = out + ((size_t)b * 64) * HW + pb;
        #pragma unroll
        for (int m = 0; m < 4; ++m) {
            #pragma unroll
            for (int r = 0; r < 8; ++r) {
                int ch = m * 16 + r + 8 * hi;
                outb[(size_t)ch * HW + pr] = acc[m][r] + sB2[ch];
            }
        }

        if (tt < 3)   // overlap next tile's global reads behind this tile's stores
            build_tile(myX, gt + 1, img, mem, midx, umem, xf, lane);
    }
}

extern "C" void kernel_launch(void* const* d_in, const int* in_sizes, int n_in,
                              void* d_out, int out_size, void* d_ws, size_t ws_size,
                              hipStream_t stream) {
    const float* img  = (const float*)d_in[0];   // (32, 64, 64, 96)
    const float* mem  = (const float*)d_in[1];   // (32, 64, 64, 96)
    const float* pext = (const float*)d_in[2];   // (32, 4, 4)
    const float* cext = (const float*)d_in[3];   // (32, 4, 4)
    const int*   midx = (const int*)d_in[4];     // (32,)
    const unsigned char* umem = (const unsigned char*)d_in[5]; // (32,) bool
    const float* Ws   = (const float*)d_in[6];   // (3, 128, 128)
    const float* bs   = (const float*)d_in[7];   // (3, 128)
    float* out = (float*)d_out;                  // (32, 64, 64, 96)
    float* xf  = (float*)d_ws;                   // 32 * 16 floats

    scrnn_xf_kernel<<<1, 32, 0, stream>>>(pext, cext, midx, umem, xf, 32);

    const size_t shmem = (size_t)SMEM_FLOATS * sizeof(float);
    // 384 blocks * 8 waves * 4 tiles * 16 px = 196608 pixels (exact cover)
    scrnn_fused_kernel<<<384, 256, shmem, stream>>>(img, mem, midx, umem, xf, Ws, bs, out);
}